// QuantumClassicalHybridNetwork_23579370455600
// MI455X (gfx1250) — compile-verified
//
#include <hip/hip_runtime.h>
#include <math.h>

#define NV 20
#define NC 64
#define NL 3
#define NSTATES (1u << 20)
#define TOPK 10

typedef __attribute__((ext_vector_type(2))) float v2f;
typedef __attribute__((ext_vector_type(8))) float v8f;

__device__ __forceinline__ float bitf(unsigned st, int v) {
    // bits[i, v] = v-th bit of i, MSB first
    return (float)((st >> (NV - 1 - v)) & 1u);
}

// ---------------------------------------------------------------------------
// Cost kernel: viol = bits @ C^T via V_WMMA_F32_16X16X4_F32.
// One wave handles 16 states x 64 constraints = 4 N-tiles x 5 K-steps = 20 WMMA.
// ---------------------------------------------------------------------------
__global__ __launch_bounds__(256) void qaoa_cost_kernel(
    const float* __restrict__ P,   // [64][20] constraints, row-major
    float* __restrict__ costs)     // [NSTATES]
{
    const int lane = threadIdx.x & 31;
    const int wave = threadIdx.x >> 5;
    const int half = lane >> 4;    // which 16-lane half of the wave
    const int l16  = lane & 15;
    const unsigned tile = blockIdx.x * 8u + (unsigned)wave;  // 65536 tiles of 16 states
    const unsigned base = tile * 16u;
    const unsigned strow = base + (unsigned)l16;             // A-matrix row (state) for this lane

    // A fragments: 16x4 f32 per K-step.
    // Lane layout (ISA 7.12.2): lanes 0-15 hold K={0,1}, lanes 16-31 hold K={2,3}.
    v2f a[5];
#pragma unroll
    for (int ks = 0; ks < 5; ++ks) {
        const int kx = ks * 4 + 2 * half;
        a[ks].x = bitf(strow, kx);
        a[ks].y = bitf(strow, kx + 1);
    }

    float csum[8];
#pragma unroll
    for (int s = 0; s < 8; ++s) csum[s] = 0.0f;

#pragma unroll
    for (int t = 0; t < 4; ++t) {            // 4 tiles of 16 constraints
        v8f acc = {0.f, 0.f, 0.f, 0.f, 0.f, 0.f, 0.f, 0.f};
        const int c = t * 16 + l16;          // B column = constraint index
        const float* prow = P + c * NV;
#pragma unroll
        for (int ks = 0; ks < 5; ++ks) {     // K = 20 = 5 x 4
            const int kb = ks * 4 + 2 * half;
            v2f b;
            b.x = prow[kb];
            b.y = prow[kb + 1];
            acc = __builtin_amdgcn_wmma_f32_16x16x4_f32(
                /*neg_a=*/false, a[ks], /*neg_b=*/false, b,
                /*c_mod=*/(short)0, acc, /*reuse_a=*/false, /*reuse_b=*/false);
        }
        // hinge^2 per element, then reduce across the 16 lanes of this half.
        // D layout: lane l<16 -> N=l, rows 0..7 in slots; lane l>=16 -> N=l-16, rows 8..15.
#pragma unroll
        for (int s = 0; s < 8; ++s) {
            float e = acc[s] - 1.0f;
            e = (e > 0.0f) ? e * e : 0.0f;
            e += __shfl_xor(e, 1, 32);
            e += __shfl_xor(e, 2, 32);
            e += __shfl_xor(e, 4, 32);
            e += __shfl_xor(e, 8, 32);       // masks <16 stay within the half
            csum[s] += e;
        }
    }

    if (l16 == 0) {                          // lane 0 -> rows 0..7, lane 16 -> rows 8..15
#pragma unroll
        for (int s = 0; s < 8; ++s) {
            const unsigned st = base + (unsigned)(half * 8 + s);
            costs[st] = csum[s] + 0.1f * (float)__popc(st);
        }
    }
}

// ---------------------------------------------------------------------------
// Elementwise / reduction pipeline kernels
// ---------------------------------------------------------------------------
__global__ __launch_bounds__(256) void qaoa_init_kernel(float* __restrict__ A) {
    const unsigned i = blockIdx.x * 256u + threadIdx.x;
    A[i] = 1.0f / 1024.0f;  // 1/sqrt(2^20)
}

__global__ __launch_bounds__(256) void qaoa_phase_kernel(
    const float* __restrict__ A, const float* __restrict__ costs,
    const float* __restrict__ gamma, int layer, float* __restrict__ B)
{
    const unsigned i = blockIdx.x * 256u + threadIdx.x;
    B[i] = A[i] * cosf(gamma[layer] * costs[i]);
}

// mixed[i] = B[i] + 0.1*sin(beta)*sum_b B[i ^ (1<<b)]; also block partial sum of mixed^2
__global__ __launch_bounds__(256) void qaoa_mix_kernel(
    const float* __restrict__ B, const float* __restrict__ beta, int layer,
    float* __restrict__ A, float* __restrict__ partials)
{
    __shared__ float red[256];
    const float sc = 0.1f * sinf(beta[layer]);
    const unsigned tid = threadIdx.x;
    float local = 0.0f;
#pragma unroll
    for (int e = 0; e < 4; ++e) {
        const unsigned i = blockIdx.x * 1024u + (unsigned)e * 256u + tid;
        float nb = 0.0f;
#pragma unroll
        for (int b = 0; b < NV; ++b) nb += B[i ^ (1u << b)];
        const float m = B[i] + sc * nb;
        A[i] = m;
        local += m * m;
    }
    red[tid] = local;
    __syncthreads();
    for (int off = 128; off > 0; off >>= 1) {
        if (tid < (unsigned)off) red[tid] += red[tid + off];
        __syncthreads();
    }
    if (tid == 0) partials[blockIdx.x] = red[0];
}

__global__ __launch_bounds__(1024) void qaoa_reduce_kernel(
    const float* __restrict__ partials, float* __restrict__ sumsq)
{
    __shared__ float red[1024];
    const unsigned tid = threadIdx.x;
    red[tid] = partials[tid];
    __syncthreads();
    for (int off = 512; off > 0; off >>= 1) {
        if (tid < (unsigned)off) red[tid] += red[tid + off];
        __syncthreads();
    }
    if (tid == 0) sumsq[0] = red[0];
}

__global__ __launch_bounds__(256) void qaoa_scale_kernel(
    float* __restrict__ A, const float* __restrict__ sumsq)
{
    const unsigned i = blockIdx.x * 256u + threadIdx.x;
    A[i] *= rsqrtf(sumsq[0]);
}

__global__ __launch_bounds__(256) void qaoa_probs_kernel(
    const float* __restrict__ A, float* __restrict__ probs)
{
    const unsigned i = blockIdx.x * 256u + threadIdx.x;
    const float a = A[i];
    probs[i] = a * a;
}

// ---------------------------------------------------------------------------
// Top-K (K=10) + final outputs. Single block, deterministic LDS argmax passes.
// Tie-break: lower index wins (matches jax.lax.top_k / argmin semantics).
// ---------------------------------------------------------------------------
__global__ __launch_bounds__(1024) void qaoa_topk_kernel(
    const float* __restrict__ probs, const float* __restrict__ costs,
    float* __restrict__ out)
{
    __shared__ float sval[1024];
    __shared__ unsigned sidx[1024];
    __shared__ unsigned sel[TOPK];
    const unsigned tid = threadIdx.x;

    for (int k = 0; k < TOPK; ++k) {
        float bv = -1.0f;
        unsigned bi = 0xFFFFFFFFu;
        for (unsigned i = tid; i < NSTATES; i += 1024u) {
            bool skip = false;
            for (int j = 0; j < k; ++j) skip = skip || (sel[j] == i);
            if (skip) continue;
            const float v = probs[i];
            if (v > bv || (v == bv && i < bi)) { bv = v; bi = i; }
        }
        sval[tid] = bv;
        sidx[tid] = bi;
        __syncthreads();
        for (int off = 512; off > 0; off >>= 1) {
            if (tid < (unsigned)off) {
                const float vo = sval[tid + off];
                const unsigned io = sidx[tid + off];
                if (vo > sval[tid] || (vo == sval[tid] && io < sidx[tid])) {
                    sval[tid] = vo; sidx[tid] = io;
                }
            }
            __syncthreads();
        }
        if (tid == 0) sel[k] = sidx[0];
        __syncthreads();
    }

    if (tid == 0) {
        const unsigned OTOPA = 21u + NSTATES;        // top_assignments [10][20]
        const unsigned OTOPC = OTOPA + TOPK * NV;    // assignment_costs [10]
        float bestc = 3.4e38f;
        int best = 0;
        for (int k = 0; k < TOPK; ++k) {
            const unsigned st = sel[k];
            const float ck = costs[st];
            out[OTOPC + k] = ck;
            for (int v = 0; v < NV; ++v) out[OTOPA + k * NV + v] = bitf(st, v);
            if (ck < bestc) { bestc = ck; best = k; }
        }
        const unsigned sb = sel[best];
        for (int v = 0; v < NV; ++v) out[v] = bitf(sb, v);  // optimal_assignment
        out[NV] = bestc;                                     // optimal_cost
    }
}

// ---------------------------------------------------------------------------
// Launch
// ---------------------------------------------------------------------------
extern "C" void kernel_launch(void* const* d_in, const int* in_sizes, int n_in,
                              void* d_out, int out_size, void* d_ws, size_t ws_size,
                              hipStream_t stream) {
    const float* P     = (const float*)d_in[0];  // [64][20]
    const float* beta  = (const float*)d_in[1];  // [3]
    const float* gamma = (const float*)d_in[2];  // [3]
    float* out = (float*)d_out;
    float* ws  = (float*)d_ws;

    float* costs    = ws;                        // [NSTATES]
    float* A        = ws + NSTATES;              // [NSTATES]
    float* B        = ws + 2u * NSTATES;         // [NSTATES]
    float* partials = ws + 3u * NSTATES;         // [1024]
    float* sumsq    = ws + 3u * NSTATES + 1024;  // [1]

    // probs live at out + 21 (after optimal_assignment[20] + optimal_cost[1])
    float* probs = out + 21;

    qaoa_cost_kernel<<<8192, 256, 0, stream>>>(P, costs);
    qaoa_init_kernel<<<NSTATES / 256, 256, 0, stream>>>(A);

    for (int layer = 0; layer < NL; ++layer) {
        qaoa_phase_kernel<<<NSTATES / 256, 256, 0, stream>>>(A, costs, gamma, layer, B);
        qaoa_mix_kernel<<<1024, 256, 0, stream>>>(B, beta, layer, A, partials);
        qaoa_reduce_kernel<<<1, 1024, 0, stream>>>(partials, sumsq);
        qaoa_scale_kernel<<<NSTATES / 256, 256, 0, stream>>>(A, sumsq);
    }

    qaoa_probs_kernel<<<NSTATES / 256, 256, 0, stream>>>(A, probs);
    qaoa_topk_kernel<<<1, 1024, 0, stream>>>(probs, costs, out);
}